// TTLinearR_91053306675883
// MI455X (gfx1250) — compile-verified
//
#include <hip/hip_runtime.h>

typedef __attribute__((ext_vector_type(2))) float v2f;
typedef __attribute__((ext_vector_type(8))) float v8f;

#define TILE_K 32

// ---------------------------------------------------------------------------
// CDNA5 async global->LDS staging (ASYNCcnt path), guarded so the file
// compiles on toolchains that don't declare the builtins.
// ---------------------------------------------------------------------------
#ifndef TT_ASYNC_LDS
#if defined(__has_builtin)
#if __has_builtin(__builtin_amdgcn_global_load_async_to_lds_b32) && \
    __has_builtin(__builtin_amdgcn_s_wait_asynccnt)
#define TT_ASYNC_LDS 1
#else
#define TT_ASYNC_LDS 0
#endif
#else
#define TT_ASYNC_LDS 0
#endif
#endif

#if TT_ASYNC_LDS
typedef __attribute__((address_space(1))) int tt_gint;   // global int
typedef __attribute__((address_space(3))) int tt_lint;   // LDS int
#endif

__device__ __forceinline__ void tt_stage_f32(const float* __restrict__ g, float* l) {
#if TT_ASYNC_LDS
    // inttoptr round-trips: AS1 keeps the full 64-bit VA; generic->AS3 is
    // low-32-bit truncation (the AMDGPU addrspacecast definition).
    __builtin_amdgcn_global_load_async_to_lds_b32(
        (tt_gint*)(unsigned long long)(const void*)g,
        (tt_lint*)(unsigned int)(unsigned long long)(void*)l,
        0, 0);
#else
    *l = *g;
#endif
}

__device__ __forceinline__ void tt_wait_stage() {
#if TT_ASYNC_LDS
    __builtin_amdgcn_s_wait_asynccnt(0);
#endif
    __syncthreads();
}

// ---------------------------------------------------------------------------
// Precompute kernels (tiny, VALU): fold the TT chain into three small factors
// ---------------------------------------------------------------------------

// Bm[r4*256 + (i4*16+i5)] = sum_r5 core4[(r4*16+i4)*64 + r5] * core5[r5*16 + i5]
__global__ __launch_bounds__(256) void tt_bm_kernel(const float* __restrict__ c4,
                                                    const float* __restrict__ c5,
                                                    float* __restrict__ Bm) {
    int idx = blockIdx.x * 256 + threadIdx.x;   // 0..16383
    int r4 = idx >> 8, c2 = idx & 255;
    int i4 = c2 >> 4, i5 = c2 & 15;
    float s = 0.f;
#pragma unroll 8
    for (int r5 = 0; r5 < 64; ++r5)
        s += c4[(r4 * 16 + i4) * 64 + r5] * c5[r5 * 16 + i5];
    Bm[idx] = s;
}

// t1[(i0*16+i1)*64 + r2] = sum_r1 core0[i0*64 + r1] * core1[(r1*16+i1)*64 + r2]
__global__ __launch_bounds__(256) void tt_t1_kernel(const float* __restrict__ c0,
                                                    const float* __restrict__ c1,
                                                    float* __restrict__ t1) {
    int idx = blockIdx.x * 256 + threadIdx.x;   // 0..16383
    int i01 = idx >> 6, r2 = idx & 63;
    int i0 = i01 >> 4, i1 = i01 & 15;
    float s = 0.f;
#pragma unroll 8
    for (int r1 = 0; r1 < 64; ++r1)
        s += c0[i0 * 64 + r1] * c1[(r1 * 16 + i1) * 64 + r2];
    t1[idx] = s;
}

// C[row*64 + r3] = sum_r2 t1[(row>>4)*64 + r2] * core2[(r2*16 + (row&15))*64 + r3]
__global__ __launch_bounds__(256) void tt_C_kernel(const float* __restrict__ t1,
                                                   const float* __restrict__ c2,
                                                   float* __restrict__ Cm) {
    int idx = blockIdx.x * 256 + threadIdx.x;   // 0..262143
    int row = idx >> 6, r3 = idx & 63;
    int i01 = row >> 4, i2 = row & 15;
    float s = 0.f;
#pragma unroll 8
    for (int r2 = 0; r2 < 64; ++r2)
        s += t1[i01 * 64 + r2] * c2[(r2 * 16 + i2) * 64 + r3];
    Cm[idx] = s;
}

// ---------------------------------------------------------------------------
// Fused steps 1+2: per 32-token block, build u[32,1024] in LDS (never touches
// HBM), then v[32,64] = u @ core3^T.
//   u[t, i3*64+r4] = sum_c2 x[t, i3*256+c2] * Bm[r4*256+c2]
//   v[t, r3]       = sum_k  u[t, k] * core3[r3*1024 + k]
// 8 waves: wm in {0,1} x wn in {0..3} of 16x16 WMMA tiles. LDS ~141 KB.
// ---------------------------------------------------------------------------
#define US_PITCH (1024 + 4)   // +4 floats: de-conflict column-strided reads

__global__ __launch_bounds__(256) void tt_uv_fused(
    const float* __restrict__ X,    // x (8192 x 4096)
    const float* __restrict__ Bm,   // (64 r4) x (256 c2)
    const float* __restrict__ G3,   // core3 viewed as (64 r3) x (1024 k)
    float* __restrict__ V) {        // v (8192 x 64)
    __shared__ float Us[32][US_PITCH];
    __shared__ float Xs[32][TILE_K + 1];
    __shared__ float Ws[64][TILE_K + 1];

    const int tid  = threadIdx.x;
    const int lane = tid & 31;          // wave32
    const int wave = tid >> 5;          // 0..7
    const int wm   = wave >> 2;         // 0..1
    const int wn   = wave & 3;          // 0..3
    const int lm   = lane & 15;
    const int kh   = lane >> 4;

    const int rowBase = blockIdx.x * 32;

    // ---- Phase A: u tile into LDS --------------------------------------
    for (int i3 = 0; i3 < 16; ++i3) {
        const float* Xp = X + i3 * 256;
        v8f acc = {};
        for (int kc = 0; kc < 256; kc += TILE_K) {
            for (int i = tid; i < 32 * TILE_K; i += 256) {
                int r = i >> 5, c = i & 31;
                tt_stage_f32(&Xp[(size_t)(rowBase + r) * 4096 + (kc + c)], &Xs[r][c]);
            }
            for (int i = tid; i < 64 * TILE_K; i += 256) {
                int r = i >> 5, c = i & 31;
                tt_stage_f32(&Bm[r * 256 + (kc + c)], &Ws[r][c]);
            }
            tt_wait_stage();
#pragma unroll
            for (int kk = 0; kk < TILE_K; kk += 4) {
                const int k0 = kk + 2 * kh;
                v2f af = { Xs[wm * 16 + lm][k0], Xs[wm * 16 + lm][k0 + 1] };
                v2f bf = { Ws[wn * 16 + lm][k0], Ws[wn * 16 + lm][k0 + 1] };
                acc = __builtin_amdgcn_wmma_f32_16x16x4_f32(
                    false, af, false, bf, (short)0, acc, false, false);
            }
            __syncthreads();
        }
        // D layout: VGPR vv -> row (vv + 8*kh), col (lane&15)
        const int ucol = i3 * 64 + wn * 16 + lm;
        const int ur0  = wm * 16 + 8 * kh;
#pragma unroll
        for (int vv = 0; vv < 8; ++vv)
            Us[ur0 + vv][ucol] = acc[vv];
    }
    __syncthreads();

    // ---- Phase B: v tile = Us(32x1024) @ core3^T(64x1024) ----------------
    v8f acc = {};
    for (int kc = 0; kc < 1024; kc += TILE_K) {
        for (int i = tid; i < 64 * TILE_K; i += 256) {
            int r = i >> 5, c = i & 31;
            tt_stage_f32(&G3[r * 1024 + (kc + c)], &Ws[r][c]);
        }
        tt_wait_stage();
#pragma unroll
        for (int kk = 0; kk < TILE_K; kk += 4) {
            const int k0 = kc + kk + 2 * kh;
            const int kw = kk + 2 * kh;
            v2f af = { Us[wm * 16 + lm][k0], Us[wm * 16 + lm][k0 + 1] };
            v2f bf = { Ws[wn * 16 + lm][kw], Ws[wn * 16 + lm][kw + 1] };
            acc = __builtin_amdgcn_wmma_f32_16x16x4_f32(
                false, af, false, bf, (short)0, acc, false, false);
        }
        __syncthreads();
    }
    const int gn = wn * 16 + lm;
    const int r0 = rowBase + wm * 16 + 8 * kh;
#pragma unroll
    for (int vv = 0; vv < 8; ++vv)
        V[(size_t)(r0 + vv) * 64 + gn] = acc[vv];
}

// ---------------------------------------------------------------------------
// WMMA fp32 GEMM:  Y[m, n] = sum_k X[m, k] * W[n, k]  (+ optional bias[n])
// Used for step 3 (store-bound on y). Block tile 32(M) x 64(N), K staged 32.
// ---------------------------------------------------------------------------
__global__ __launch_bounds__(256) void tt_gemm_wmma(
    const float* __restrict__ X, int ldx,
    const float* __restrict__ W,                       // N x K row-major
    float* __restrict__ Y, int ldy,
    int K, const float* __restrict__ bias) {
    __shared__ float Xs[32][TILE_K + 1];
    __shared__ float Ws[64][TILE_K + 1];

    const int tid  = threadIdx.x;
    const int lane = tid & 31;
    const int wave = tid >> 5;
    const int wm   = wave >> 2;
    const int wn   = wave & 3;
    const int lm   = lane & 15;
    const int kh   = lane >> 4;

    const int rowBase = blockIdx.x * 32;
    const int nBase   = blockIdx.y * 64;

    v8f acc = {};

    for (int kc = 0; kc < K; kc += TILE_K) {
        for (int i = tid; i < 32 * TILE_K; i += 256) {
            int r = i >> 5, c = i & 31;
            tt_stage_f32(&X[(size_t)(rowBase + r) * ldx + (kc + c)], &Xs[r][c]);
        }
        for (int i = tid; i < 64 * TILE_K; i += 256) {
            int r = i >> 5, c = i & 31;
            tt_stage_f32(&W[(size_t)(nBase + r) * K + (kc + c)], &Ws[r][c]);
        }
        tt_wait_stage();
#pragma unroll
        for (int kk = 0; kk < TILE_K; kk += 4) {
            const int k0 = kk + 2 * kh;
            v2f af = { Xs[wm * 16 + lm][k0], Xs[wm * 16 + lm][k0 + 1] };
            v2f bf = { Ws[wn * 16 + lm][k0], Ws[wn * 16 + lm][k0 + 1] };
            acc = __builtin_amdgcn_wmma_f32_16x16x4_f32(
                false, af, false, bf, (short)0, acc, false, false);
        }
        __syncthreads();
    }

    const int gn = nBase + wn * 16 + lm;
    const float bv = bias ? bias[gn] : 0.f;
    const int r0 = rowBase + wm * 16 + 8 * kh;
#pragma unroll
    for (int vv = 0; vv < 8; ++vv)
        Y[(size_t)(r0 + vv) * ldy + gn] = acc[vv] + bv;
}

// ---------------------------------------------------------------------------
// y = x @ W(TT)^T + b without materializing W: ~9.7 GFLOP, ~258 MB HBM
// traffic (x 128 MB + y 128 MB + v 2 MB) -> ~11 us at 23.3 TB/s.
// ---------------------------------------------------------------------------
extern "C" void kernel_launch(void* const* d_in, const int* in_sizes, int n_in,
                              void* d_out, int out_size, void* d_ws, size_t ws_size,
                              hipStream_t stream) {
    const float* c0 = (const float*)d_in[0];   // (1,16,64)
    const float* c1 = (const float*)d_in[1];   // (64,16,64)
    const float* c2 = (const float*)d_in[2];   // (64,16,64)
    const float* c3 = (const float*)d_in[3];   // (64,16,64) == (64 x 1024)
    const float* c4 = (const float*)d_in[4];   // (64,16,64)
    const float* c5 = (const float*)d_in[5];   // (64,16,1)
    const float* x  = (const float*)d_in[6];   // (8192, 4096)
    const float* bb = (const float*)d_in[7];   // (4096)
    float* y = (float*)d_out;                  // (8192, 4096)

    // Workspace (fp32 elements): ~3.2 MB
    float* ws = (float*)d_ws;
    float* Bm = ws;                     //  16384 : (64 r4) x (256 c2)
    float* t1 = ws + 16384;             //  16384 : (256 i0i1) x (64 r2)
    float* Cm = ws + 32768;             // 262144 : (4096 out) x (64 r3)
    float* vv = ws + 294912;            // 524288 : (8192 t) x (64 r3)

    // Fold TT cores into the three small factors
    tt_bm_kernel<<<64,   256, 0, stream>>>(c4, c5, Bm);
    tt_t1_kernel<<<64,   256, 0, stream>>>(c0, c1, t1);
    tt_C_kernel <<<1024, 256, 0, stream>>>(t1, c2, Cm);

    // Steps 1+2 fused: v[t, r3] from x, with u tile resident in LDS
    tt_uv_fused<<<256, 256, 0, stream>>>(x, Bm, c3, vv);

    // Step 3: y[t, p] = sum_r3 v[t, r3] * C[p*64 + r3] + b[p]
    tt_gemm_wmma<<<dim3(256, 64, 1), 256, 0, stream>>>(
        vv, 64, Cm, y, 4096, /*K=*/64, bb);
}